// GDS_2130303779115
// MI455X (gfx1250) — compile-verified
//
#include <hip/hip_runtime.h>

typedef __attribute__((ext_vector_type(2))) float v2f;
typedef __attribute__((ext_vector_type(8))) float v8f;

// ---------------------------------------------------------------------------
// DIM = DEG = 8, NEU = 4 (fixed by the reference).
// Terms n = 5..8 use kron^n(x) = kron^{n-4}(x) (x) kron^4(x):
//   out[nn][p] += sum_hi K_{n-4}[nn][hi] * sum_lo c_n[p][hi*4096+lo]*K4[nn][lo]
// Work unit = one wave: 16 coefficient rows (2 hi x 8 p) x 1024 lo values,
// reduced with V_WMMA_F32_16X16X4_F32 (A = c rows, B = K4, N = neuron).
//   c5:16  c6:128  c7:1024  c8:8192  -> 9360 units = 1170 blocks * 8 waves.
// ---------------------------------------------------------------------------
#define TOTAL_UNITS 9360
#define MAIN_BLOCKS 1170

// ---------------------------------------------------------------------------
// Build K4[neuron][lo], lo in [0,4096), pair-interleaved for conflict-free
// ds_load_b64 B-fragments:  ws[(lo>>1)*8 + n*2 + (lo&1)].
__global__ void setup_k4_kernel(const float* __restrict__ X,
                                float* __restrict__ wsK4) {
    for (int lo = threadIdx.x; lo < 4096; lo += blockDim.x) {
        int a = (lo >> 9) & 7, b = (lo >> 6) & 7, c = (lo >> 3) & 7, d = lo & 7;
        #pragma unroll
        for (int n = 0; n < 4; ++n) {
            float v = X[n * 8 + a] * X[n * 8 + b] * X[n * 8 + c] * X[n * 8 + d];
            wsK4[(lo >> 1) * 8 + n * 2 + (lo & 1)] = v;
        }
    }
}

// ---------------------------------------------------------------------------
// Terms n = 1..4 (~150 KB total): plain VALU, one block, deterministic
// fixed-order reduction into a 32-float partial in d_ws.
__global__ __launch_bounds__(256) void small_terms_kernel(
    const float* __restrict__ X,
    const float* __restrict__ c1, const float* __restrict__ c2,
    const float* __restrict__ c3, const float* __restrict__ c4,
    float* __restrict__ wsSmall) {
    __shared__ float sBuf[256 * 32];   // 32 KB
    float acc[32];
    #pragma unroll
    for (int i = 0; i < 32; ++i) acc[i] = 0.0f;

    for (int idx = threadIdx.x; idx < 4680; idx += 256) {
        int m, k, S;
        const float* cm;
        if (idx < 8)        { m = 1; k = idx;       cm = c1; S = 8;    }
        else if (idx < 72)  { m = 2; k = idx - 8;   cm = c2; S = 64;   }
        else if (idx < 584) { m = 3; k = idx - 72;  cm = c3; S = 512;  }
        else                { m = 4; k = idx - 584; cm = c4; S = 4096; }

        float Kv[4];
        #pragma unroll
        for (int n = 0; n < 4; ++n) {
            float v = 1.0f;
            for (int d = 0; d < m; ++d) v *= X[n * 8 + ((k >> (3 * d)) & 7)];
            Kv[n] = v;
        }
        #pragma unroll
        for (int p = 0; p < 8; ++p) {
            float cv = cm[p * S + k];
            #pragma unroll
            for (int n = 0; n < 4; ++n) acc[n * 8 + p] += cv * Kv[n];
        }
    }
    #pragma unroll
    for (int i = 0; i < 32; ++i) sBuf[threadIdx.x * 32 + i] = acc[i];
    __syncthreads();
    if (threadIdx.x < 32) {
        float s = 0.0f;
        for (int t = 0; t < 256; ++t) s += sBuf[t * 32 + threadIdx.x];
        wsSmall[threadIdx.x] = s;
    }
}

// ---------------------------------------------------------------------------
// Main streaming WMMA kernel: each wave owns exactly one work unit.
// Per-block deterministic LDS reduction (reusing sK4 after a barrier),
// one 32-float partial written per block -> no global float atomics.
__global__ __launch_bounds__(256) void poly_wmma_kernel(
    const float* __restrict__ X,
    const float* __restrict__ c5, const float* __restrict__ c6,
    const float* __restrict__ c7, const float* __restrict__ c8,
    const float* __restrict__ wsK4, float* __restrict__ wsPart) {
    // K4 table, pair-interleaved: sK4[(lo>>1)*8 + n*2 + (lo&1)]  (exactly 64 KB)
    __shared__ float sK4[16384];
    for (int i = threadIdx.x; i < 16384; i += blockDim.x) sK4[i] = wsK4[i];
    __syncthreads();

    const int lane  = threadIdx.x & 31;
    const int Lm    = lane & 15;     // A-row / B-column (N = neuron) index
    const int isHi  = lane >> 4;     // lane half selects K offset {0,1} vs {2,3}
    const int n4    = Lm & 3;
    const int wave  = threadIdx.x >> 5;
    const float msk = (Lm < 4) ? 1.0f : 0.0f;   // zero B columns 4..15

    const int unit = blockIdx.x * 8 + wave;     // grid sized so unit < TOTAL_UNITS

    int nt, tu;
    const float* cbase;
    long strideN;
    if (unit < 16)        { nt = 5; tu = unit;        cbase = c5; strideN = 32768L;    }
    else if (unit < 144)  { nt = 6; tu = unit - 16;   cbase = c6; strideN = 262144L;   }
    else if (unit < 1168) { nt = 7; tu = unit - 144;  cbase = c7; strideN = 2097152L;  }
    else                  { nt = 8; tu = unit - 1168; cbase = c8; strideN = 16777216L; }
    const int tile = tu >> 2;
    const int seg  = tu & 3;

    // This lane's A row: r = (hi, p); row data contiguous along lo.
    const int r  = tile * 16 + Lm;
    const int hi = r >> 3;
    const int p  = r & 7;
    (void)p;

    const float* gA = cbase + (long)p * strideN + (long)hi * 4096
                    + (long)(seg * 1024 + 2 * isHi);
    const float* lB = sK4 + ((seg * 512 + isHi) * 8 + n4 * 2);

    v8f acc = {0.f, 0.f, 0.f, 0.f, 0.f, 0.f, 0.f, 0.f};

    // 256 WMMAs reduce the 1024-long lo segment; unroll 16 => 32 cachelines
    // outstanding per wave for HBM saturation (Little's law ~14MB in flight).
    #pragma unroll 16
    for (int k = 0; k < 1024; k += 4) {
        v2f a = *(const v2f*)(gA + k);              // A: c rows (global_load_b64)
        v2f b = *(const v2f*)(lB + (k >> 1) * 8);   // B: K4     (ds_load_b64)
        b.x *= msk;
        b.y *= msk;
        acc = __builtin_amdgcn_wmma_f32_16x16x4_f32(
            false, a, false, b, (short)0, acc, false, false);
    }

    // hh = ((tile*16 + v + 8*isHi) >> 3) = tile*2 + isHi for all v, and p = v:
    // compute K_{nt-4}[neuron][hh] once per unit.
    const int hh   = tile * 2 + isHi;
    const int digs = nt - 4;
    float khi = 1.0f;
    for (int d = 0; d < digs; ++d)
        khi *= X[n4 * 8 + ((hh >> (3 * d)) & 7)];

    // Deterministic block reduction, reusing sK4 (all waves done reading it).
    __syncthreads();
    if (Lm < 4) {
        const int slot = ((wave * 2 + isHi) * 4 + n4) * 8;
        #pragma unroll
        for (int v = 0; v < 8; ++v) sK4[slot + v] = acc[v] * khi;
    }
    __syncthreads();
    if (threadIdx.x < 32) {
        const int n = threadIdx.x >> 3;
        const int pp = threadIdx.x & 7;
        float s = 0.0f;
        #pragma unroll
        for (int w = 0; w < 16; ++w) s += sK4[(w * 4 + n) * 8 + pp];
        wsPart[blockIdx.x * 32 + threadIdx.x] = s;
    }
}

// ---------------------------------------------------------------------------
// Final fixed-order reduction: out = bias + small terms + sum of block partials.
__global__ __launch_bounds__(256) void finish_kernel(
    const float* __restrict__ bias, const float* __restrict__ wsPart,
    const float* __restrict__ wsSmall, float* __restrict__ out) {
    __shared__ float sT[256];
    const int t    = threadIdx.x;
    const int oidx = t & 31;
    const int j    = t >> 5;           // 8 accumulator lanes per output
    float s = 0.0f;
    for (int b = j; b < MAIN_BLOCKS; b += 8) s += wsPart[b * 32 + oidx];
    sT[t] = s;
    __syncthreads();
    if (t < 32) {
        float tot = bias[t & 7] + wsSmall[t];
        #pragma unroll
        for (int jj = 0; jj < 8; ++jj) tot += sT[jj * 32 + t];
        out[t] = tot;
    }
}

// ---------------------------------------------------------------------------
extern "C" void kernel_launch(void* const* d_in, const int* in_sizes, int n_in,
                              void* d_out, int out_size, void* d_ws, size_t ws_size,
                              hipStream_t stream) {
    const float* X    = (const float*)d_in[0];
    const float* bias = (const float*)d_in[1];
    const float* c1   = (const float*)d_in[2];
    const float* c2   = (const float*)d_in[3];
    const float* c3   = (const float*)d_in[4];
    const float* c4   = (const float*)d_in[5];
    const float* c5   = (const float*)d_in[6];
    const float* c6   = (const float*)d_in[7];
    const float* c7   = (const float*)d_in[8];
    const float* c8   = (const float*)d_in[9];
    float* out = (float*)d_out;

    float* wsK4    = (float*)d_ws;                   // 16384 floats
    float* wsPart  = wsK4 + 16384;                   // 1170*32 floats
    float* wsSmall = wsPart + MAIN_BLOCKS * 32;      // 32 floats  (~215 KB total)

    setup_k4_kernel<<<1, 256, 0, stream>>>(X, wsK4);
    small_terms_kernel<<<1, 256, 0, stream>>>(X, c1, c2, c3, c4, wsSmall);
    poly_wmma_kernel<<<MAIN_BLOCKS, 256, 0, stream>>>(X, c5, c6, c7, c8, wsK4, wsPart);
    finish_kernel<<<1, 256, 0, stream>>>(bias, wsPart, wsSmall, out);
}